// GIN_15616501088825
// MI455X (gfx1250) — compile-verified
//
#include <hip/hip_runtime.h>
#include <math.h>

#define D 64

typedef __attribute__((ext_vector_type(2))) float v2f;
typedef __attribute__((ext_vector_type(8))) float v8f;

// ---------------------------------------------------------------------------
// Kernel 1: agg = x   (since h = (1+eps)*x + sum_neighbors, eps = 0)
// ---------------------------------------------------------------------------
__global__ void gin_init_agg(const float4* __restrict__ x,
                             float4* __restrict__ agg, int n4) {
  int t = blockIdx.x * blockDim.x + threadIdx.x;
  if (t < n4) agg[t] = x[t];
}

// ---------------------------------------------------------------------------
// Kernel 2: edge scatter-add.  One thread handles a float4 chunk of one edge:
//   agg[dst[e]][c..c+3] += x[src[e]][c..c+3]
// float4 gather (global_load_b128) + 4x global_atomic_add_f32 (lands in L2;
// both tables fit in the 192 MB L2).
// ---------------------------------------------------------------------------
__global__ void gin_scatter(const float* __restrict__ x,
                            const int* __restrict__ src,
                            const int* __restrict__ dst,
                            float* __restrict__ agg, int E) {
  int t = blockIdx.x * blockDim.x + threadIdx.x;
  int e = t >> 4;                 // 16 threads per edge (16 * float4 = 64 feats)
  if (e >= E) return;
  int c = (t & 15) << 2;
  int s = src[e];
  int d = dst[e];
  float4 v = *(const float4*)(x + (size_t)s * D + c);
  float* a = agg + (size_t)d * D + c;
  atomicAdd(a + 0, v.x);
  atomicAdd(a + 1, v.y);
  atomicAdd(a + 2, v.z);
  atomicAdd(a + 3, v.w);
}

// ---------------------------------------------------------------------------
// Kernel 3: out = tanh(agg @ W^T + b) using V_WMMA_F32_16X16X4_F32 (exact f32).
// One wave32 per 16-row block, 4 column tiles (64 outputs) per wave.
// W staged in LDS (16 KB); K-loop of 16 steps -> 64 WMMA ops per wave.
// ---------------------------------------------------------------------------
__global__ __launch_bounds__(256) void gin_linear_tanh(
    const float* __restrict__ agg, const float* __restrict__ W,
    const float* __restrict__ bias, float* __restrict__ out, int N) {
  __shared__ float Ws[D * D];     // W[out][in] row-major, 16 KB

  // Cooperative stage of W into LDS (1024 float4 / 256 threads)
  {
    const float4* Wv = (const float4*)W;
    float4* Sv = (float4*)Ws;
#pragma unroll
    for (int i = 0; i < 4; ++i) {
      int idx = threadIdx.x + i * 256;
      Sv[idx] = Wv[idx];
    }
  }
  __syncthreads();

  const int wave = threadIdx.x >> 5;
  const int lane = threadIdx.x & 31;
  const int rowblock = blockIdx.x * 8 + wave;
  const int nblocks = (N + 15) >> 4;
  if (rowblock >= nblocks) return;      // wave-uniform exit; EXEC stays full

  const int rowbase = rowblock << 4;
  const int half = (lane >= 16) ? 1 : 0;   // half-wave selects K pair
  const int l15 = lane & 15;

  // A fragment source row (clamped so loads never diverge; N % 16 == 0 anyway)
  int arow = rowbase + l15;
  if (arow > N - 1) arow = N - 1;
  const float* __restrict__ arow_p = agg + (size_t)arow * D;

  v8f acc0 = {}, acc1 = {}, acc2 = {}, acc3 = {};

#pragma unroll
  for (int k0 = 0; k0 < D; k0 += 4) {
    const int k = k0 + 2 * half;
    // A (16x4, MxK): lane = row M, VGPR0/1 = K=k, k+1 (half-wave -> k+2,k+3)
    v2f a;
    a.x = arow_p[k];
    a.y = arow_p[k + 1];
    // B (4x16, KxN): B[k][n] = W^T[k][n] = W[n][k]; lane = col n
    const float* w0 = Ws + (0 * 16 + l15) * D + k;
    const float* w1 = Ws + (1 * 16 + l15) * D + k;
    const float* w2 = Ws + (2 * 16 + l15) * D + k;
    const float* w3 = Ws + (3 * 16 + l15) * D + k;
    v2f b0; b0.x = w0[0]; b0.y = w0[1];
    v2f b1; b1.x = w1[0]; b1.y = w1[1];
    v2f b2; b2.x = w2[0]; b2.y = w2[1];
    v2f b3; b3.x = w3[0]; b3.y = w3[1];

    acc0 = __builtin_amdgcn_wmma_f32_16x16x4_f32(false, a, false, b0, (short)0, acc0, false, false);
    acc1 = __builtin_amdgcn_wmma_f32_16x16x4_f32(false, a, false, b1, (short)0, acc1, false, false);
    acc2 = __builtin_amdgcn_wmma_f32_16x16x4_f32(false, a, false, b2, (short)0, acc2, false, false);
    acc3 = __builtin_amdgcn_wmma_f32_16x16x4_f32(false, a, false, b3, (short)0, acc3, false, false);
  }

  // Epilogue: bias + tanh + store.
  const float bb0 = bias[0 * 16 + l15];
  const float bb1 = bias[1 * 16 + l15];
  const float bb2 = bias[2 * 16 + l15];
  const float bb3 = bias[3 * 16 + l15];

#pragma unroll
  for (int r = 0; r < 8; ++r) {
    int row = rowbase + r + 8 * half;   // C/D layout: M = r + 8*(lane>=16)
    if (row < N) {
      float* o = out + (size_t)row * D;
      o[0 * 16 + l15] = tanhf(acc0[r] + bb0);
      o[1 * 16 + l15] = tanhf(acc1[r] + bb1);
      o[2 * 16 + l15] = tanhf(acc2[r] + bb2);
      o[3 * 16 + l15] = tanhf(acc3[r] + bb3);
    }
  }
}

// ---------------------------------------------------------------------------
extern "C" void kernel_launch(void* const* d_in, const int* in_sizes, int n_in,
                              void* d_out, int out_size, void* d_ws, size_t ws_size,
                              hipStream_t stream) {
  const float* x   = (const float*)d_in[0];   // [N, 64]
  const float* W   = (const float*)d_in[1];   // [64, 64]
  const float* b   = (const float*)d_in[2];   // [64]
  const int*   src = (const int*)d_in[3];     // [E]
  const int*   dst = (const int*)d_in[4];     // [E]
  float* out = (float*)d_out;                 // [N, 64]
  float* agg = (float*)d_ws;                  // [N, 64] scratch (12.8 MB)

  const int N = in_sizes[0] / D;
  const int E = in_sizes[3];

  // 1) agg = x
  const int n4 = N * D / 4;
  gin_init_agg<<<(n4 + 255) / 256, 256, 0, stream>>>(
      (const float4*)x, (float4*)agg, n4);

  // 2) agg[dst] += x[src]
  const long long sthreads = (long long)E * 16;
  gin_scatter<<<(unsigned)((sthreads + 255) / 256), 256, 0, stream>>>(
      x, src, dst, agg, E);

  // 3) out = tanh(agg @ W^T + b) via f32 WMMA
  const int nblocks16 = (N + 15) / 16;
  gin_linear_tanh<<<(nblocks16 + 7) / 8, 256, 0, stream>>>(
      agg, W, b, out, N);
}